// GATBlock_66932770341398
// MI455X (gfx1250) — compile-verified
//
#include <hip/hip_runtime.h>
#include <hip/hip_bf16.h>
#include <math.h>

typedef float v2f __attribute__((ext_vector_type(2)));
typedef float v8f __attribute__((ext_vector_type(8)));

#define HEADS 4
#define HID 128
#define HD 512           // HEADS * HID
#define NEG_SLOPE 0.2f
#define BN_EPS 1e-5f

// ---------------- float atomic max (bit trick) ----------------
__device__ __forceinline__ void atomicMaxF(float* addr, float value) {
    if (value >= 0.0f) {
        atomicMax((int*)addr, __float_as_int(value));
    } else {
        atomicMin((unsigned int*)addr, (unsigned int)__float_as_int(value));
    }
}

// ---------------- init workspace ----------------
__global__ void init_ws_kernel(float* emax, float* denom, float* outacc,
                               float* stats, int N) {
    int i = blockIdx.x * 256 + threadIdx.x;
    int n4 = N * 4;
    int nacc = N * HID;
    int total = n4 + n4 + nacc + 256;
    if (i >= total) return;
    if (i < n4) {
        emax[i] = __int_as_float(0xFF800000); // -inf
    } else if (i < 2 * n4) {
        denom[i - n4] = 0.0f;
    } else if (i < 2 * n4 + nacc) {
        outacc[i - 2 * n4] = 0.0f;
    } else {
        stats[i - 2 * n4 - nacc] = 0.0f;
    }
}

// ---------------- GEMM: out[N,512] = x[N,128] @ W[128,512] + b ----------------
// One block = 8 waves; block handles a 16-row strip; each wave handles 64 cols.
// Uses V_WMMA_F32_16X16X4_F32, K stepped by 4 over the 128-deep reduction.
__global__ __launch_bounds__(256) void gemm_wmma_kernel(
    const float* __restrict__ x, const float* __restrict__ W,
    const float* __restrict__ b, float* __restrict__ out, int N) {
    __shared__ float lds_x[16 * 128];

    int m0 = blockIdx.x * 16;
    // cooperative load of 16x128 x tile (8 KB)
    for (int i = threadIdx.x; i < 16 * 128; i += 256) {
        int row = m0 + (i >> 7);
        lds_x[i] = (row < N) ? x[(size_t)row * 128 + (i & 127)] : 0.0f;
    }
    __syncthreads();

    int wave = threadIdx.x >> 5;   // 0..7 -> column group of 64
    int lane = threadIdx.x & 31;
    int n0 = wave * 64;
    int col = lane & 15;
    int khalf = lane >> 4;         // lanes 0-15: K{0,1}; lanes 16-31: K{2,3}

    v8f acc0 = {}, acc1 = {}, acc2 = {}, acc3 = {};

#pragma unroll 4
    for (int k = 0; k < 128; k += 4) {
        v2f a;
        a.x = lds_x[col * 128 + k + 2 * khalf];
        a.y = lds_x[col * 128 + k + 2 * khalf + 1];
        const float* Wrow0 = W + (size_t)(k + 2 * khalf) * HD;
        const float* Wrow1 = Wrow0 + HD;

        v2f b0; b0.x = Wrow0[n0 + 0 * 16 + col];  b0.y = Wrow1[n0 + 0 * 16 + col];
        v2f b1; b1.x = Wrow0[n0 + 1 * 16 + col];  b1.y = Wrow1[n0 + 1 * 16 + col];
        v2f b2; b2.x = Wrow0[n0 + 2 * 16 + col];  b2.y = Wrow1[n0 + 2 * 16 + col];
        v2f b3; b3.x = Wrow0[n0 + 3 * 16 + col];  b3.y = Wrow1[n0 + 3 * 16 + col];

        acc0 = __builtin_amdgcn_wmma_f32_16x16x4_f32(false, a, false, b0, (short)0, acc0, false, false);
        acc1 = __builtin_amdgcn_wmma_f32_16x16x4_f32(false, a, false, b1, (short)0, acc1, false, false);
        acc2 = __builtin_amdgcn_wmma_f32_16x16x4_f32(false, a, false, b2, (short)0, acc2, false, false);
        acc3 = __builtin_amdgcn_wmma_f32_16x16x4_f32(false, a, false, b3, (short)0, acc3, false, false);
    }

    v8f* accs[4] = {&acc0, &acc1, &acc2, &acc3};
#pragma unroll
    for (int t = 0; t < 4; ++t) {
        int nn = n0 + t * 16 + col;
        float bv = b[nn];
        v8f a = *accs[t];
#pragma unroll
        for (int i = 0; i < 8; ++i) {
            int m = m0 + i + 8 * khalf;
            if (m < N) out[(size_t)m * HD + nn] = a[i] + bv;
        }
    }
}

// ---------------- edge scores: e[E',H], atomic max into emax[N,H] ----------------
// One wave per edge; lane L covers elements [L*16, L*16+16); head = L/8.
__global__ __launch_bounds__(256) void edge_score_kernel(
    const int* __restrict__ srcIdx, const int* __restrict__ dstIdx,
    const float* __restrict__ xl, const float* __restrict__ xr,
    const float* __restrict__ att, float* __restrict__ e,
    float* __restrict__ emax, int E, int Ep) {
    int wave = threadIdx.x >> 5;
    int lane = threadIdx.x & 31;
    int eid = blockIdx.x * 8 + wave;
    if (eid >= Ep) return;
    int s, d;
    if (eid < E) { s = srcIdx[eid]; d = dstIdx[eid]; }
    else { s = eid - E; d = eid - E; }

    const float4* pl = (const float4*)(xl + (size_t)s * HD + lane * 16);
    const float4* pr = (const float4*)(xr + (size_t)d * HD + lane * 16);
    const float4* pa = (const float4*)(att + lane * 16);

    float acc = 0.0f;
#pragma unroll
    for (int q = 0; q < 4; ++q) {
        float4 a4 = pl[q], r4 = pr[q], w4 = pa[q];
        float v;
        v = a4.x + r4.x; v = v > 0.0f ? v : NEG_SLOPE * v; acc += v * w4.x;
        v = a4.y + r4.y; v = v > 0.0f ? v : NEG_SLOPE * v; acc += v * w4.y;
        v = a4.z + r4.z; v = v > 0.0f ? v : NEG_SLOPE * v; acc += v * w4.z;
        v = a4.w + r4.w; v = v > 0.0f ? v : NEG_SLOPE * v; acc += v * w4.w;
    }
    // reduce within each 8-lane group (one head)
    acc += __shfl_xor(acc, 1, 32);
    acc += __shfl_xor(acc, 2, 32);
    acc += __shfl_xor(acc, 4, 32);
    if ((lane & 7) == 0) {
        int h = lane >> 3;
        e[(size_t)eid * HEADS + h] = acc;
        atomicMaxF(&emax[d * HEADS + h], acc);
    }
}

// ---------------- exp + denominator ----------------
__global__ void edge_exp_kernel(const int* __restrict__ dstIdx,
                                const float* __restrict__ emax,
                                float* __restrict__ e,
                                float* __restrict__ denom, int E, int Ep) {
    int i = blockIdx.x * 256 + threadIdx.x;
    if (i >= Ep * HEADS) return;
    int eid = i >> 2;
    int h = i & 3;
    int d = (eid < E) ? dstIdx[eid] : (eid - E);
    float ee = __expf(e[i] - emax[d * HEADS + h]);
    e[i] = ee;  // overwrite with exp value
    atomicAdd(&denom[d * HEADS + h], ee);
}

// ---------------- aggregate: outacc[dst,d] += sum_h alpha_h * xl[src,h,d] ----------------
// One wave per edge; lane L covers d = L*4..L*4+3, all 4 heads fused -> 1 atomic per (edge,d).
__global__ __launch_bounds__(256) void edge_agg_kernel(
    const int* __restrict__ srcIdx, const int* __restrict__ dstIdx,
    const float* __restrict__ xl, const float* __restrict__ ee,
    const float* __restrict__ denom, float* __restrict__ outacc,
    int E, int Ep) {
    int wave = threadIdx.x >> 5;
    int lane = threadIdx.x & 31;
    int eid = blockIdx.x * 8 + wave;
    if (eid >= Ep) return;
    int s, d;
    if (eid < E) { s = srcIdx[eid]; d = dstIdx[eid]; }
    else { s = eid - E; d = eid - E; }

    float alpha[HEADS];
#pragma unroll
    for (int h = 0; h < HEADS; ++h) {
        alpha[h] = ee[(size_t)eid * HEADS + h] / (denom[d * HEADS + h] + 1e-16f);
    }
    float ax = 0.0f, ay = 0.0f, az = 0.0f, aw = 0.0f;
#pragma unroll
    for (int h = 0; h < HEADS; ++h) {
        float4 v = *(const float4*)(xl + (size_t)s * HD + h * HID + lane * 4);
        ax += alpha[h] * v.x;
        ay += alpha[h] * v.y;
        az += alpha[h] * v.z;
        aw += alpha[h] * v.w;
    }
    float* o = outacc + (size_t)d * HID + lane * 4;
    atomicAdd(o + 0, ax);
    atomicAdd(o + 1, ay);
    atomicAdd(o + 2, az);
    atomicAdd(o + 3, aw);
}

// ---------------- head-mean + bias, accumulate BN statistics ----------------
__global__ __launch_bounds__(256) void bn_stats_kernel(
    float* __restrict__ outacc, const float* __restrict__ bias,
    float* __restrict__ stats, int N) {
    __shared__ float sh[256], sh2[256];
    int dcol = threadIdx.x & 127;
    int rhalf = threadIdx.x >> 7;
    int n0 = blockIdx.x * 64;
    float s = 0.0f, s2 = 0.0f;
    for (int r = rhalf; r < 64; r += 2) {
        int n = n0 + r;
        if (n < N) {
            size_t idx = (size_t)n * HID + dcol;
            float v = outacc[idx] * 0.25f + bias[dcol];
            outacc[idx] = v;
            s += v;
            s2 += v * v;
        }
    }
    sh[threadIdx.x] = s;
    sh2[threadIdx.x] = s2;
    __syncthreads();
    if (rhalf == 0) {
        atomicAdd(&stats[dcol], s + sh[threadIdx.x + 128]);
        atomicAdd(&stats[128 + dcol], s2 + sh2[threadIdx.x + 128]);
    }
}

// ---------------- BatchNorm finalize + ReLU ----------------
__global__ void bn_final_kernel(const float* __restrict__ outv,
                                const float* __restrict__ stats,
                                const float* __restrict__ gamma,
                                const float* __restrict__ beta,
                                float* __restrict__ y, int N) {
    int i = blockIdx.x * 256 + threadIdx.x;
    if (i >= N * HID) return;
    int d = i & 127;
    float invN = 1.0f / (float)N;
    float mu = stats[d] * invN;
    float var = stats[128 + d] * invN - mu * mu;
    float v = gamma[d] * (outv[i] - mu) * rsqrtf(var + BN_EPS) + beta[d];
    y[i] = v > 0.0f ? v : 0.0f;
}

extern "C" void kernel_launch(void* const* d_in, const int* in_sizes, int n_in,
                              void* d_out, int out_size, void* d_ws, size_t ws_size,
                              hipStream_t stream) {
    const float* x     = (const float*)d_in[0];
    const int*   eidx  = (const int*)d_in[1];
    const float* W_l   = (const float*)d_in[2];
    const float* b_l   = (const float*)d_in[3];
    const float* W_r   = (const float*)d_in[4];
    const float* b_r   = (const float*)d_in[5];
    const float* att   = (const float*)d_in[6];
    const float* bias  = (const float*)d_in[7];
    const float* gamma = (const float*)d_in[8];
    const float* beta  = (const float*)d_in[9];
    float* y = (float*)d_out;

    int N  = in_sizes[0] / HID;       // 10000
    int E  = in_sizes[1] / 2;         // 320000
    int Ep = E + N;                   // with self loops

    const int* srcIdx = eidx;         // edge_index[0], first E entries
    const int* dstIdx = eidx + E;     // edge_index[1]

    float* ws = (float*)d_ws;
    size_t off = 0;
    float* xl     = ws + off; off += (size_t)N * HD;     // [N,512]
    float* xr     = ws + off; off += (size_t)N * HD;     // [N,512]
    float* e      = ws + off; off += (size_t)Ep * HEADS; // [E',4] scores -> exp
    float* emax   = ws + off; off += (size_t)N * HEADS;  // [N,4]
    float* denom  = ws + off; off += (size_t)N * HEADS;  // [N,4]
    float* outacc = ws + off; off += (size_t)N * HID;    // [N,128]
    float* stats  = ws + off; off += 256;                // sum[128], sumsq[128]

    // 1. init emax/denom/outacc/stats
    {
        int total = N * 4 * 2 + N * HID + 256;
        init_ws_kernel<<<(total + 255) / 256, 256, 0, stream>>>(emax, denom, outacc, stats, N);
    }
    // 2/3. GEMMs (WMMA f32 16x16x4)
    {
        int grid = (N + 15) / 16;
        gemm_wmma_kernel<<<grid, 256, 0, stream>>>(x, W_l, b_l, xl, N);
        gemm_wmma_kernel<<<grid, 256, 0, stream>>>(x, W_r, b_r, xr, N);
    }
    // 4. per-edge attention scores + segment max
    edge_score_kernel<<<(Ep + 7) / 8, 256, 0, stream>>>(srcIdx, dstIdx, xl, xr, att, e, emax, E, Ep);
    // 5. exp + denominator
    edge_exp_kernel<<<(Ep * HEADS + 255) / 256, 256, 0, stream>>>(dstIdx, emax, e, denom, E, Ep);
    // 6. weighted aggregation (head-fused)
    edge_agg_kernel<<<(Ep + 7) / 8, 256, 0, stream>>>(srcIdx, dstIdx, xl, e, denom, outacc, E, Ep);
    // 7. head-mean + bias + BN stats
    bn_stats_kernel<<<(N + 63) / 64, 256, 0, stream>>>(outacc, bias, stats, N);
    // 8. BN normalize + ReLU
    bn_final_kernel<<<((size_t)N * HID + 255) / 256, 256, 0, stream>>>(outacc, stats, gamma, beta, y, N);
}